// CRF_79663053406323
// MI455X (gfx1250) — compile-verified
//
#include <hip/hip_runtime.h>
#include <hip/hip_bf16.h>

#define B_ 32
#define S_ 512
#define H_ 256
#define T_ 48
#define START_IDX 46
#define STOP_IDX 47
#define NEGV (-10000.0f)

typedef __attribute__((ext_vector_type(2))) float v2f;
typedef __attribute__((ext_vector_type(8))) float v8f;

#if __has_builtin(__builtin_amdgcn_wmma_f32_16x16x4_f32)
#define HAVE_WMMA_F32 1
#else
#define HAVE_WMMA_F32 0
#endif

// Fused CRF NLL kernel: one workgroup per batch element.
// Phase A: emission GEMM (512x256 @ 256x48) -> em[] in LDS via V_WMMA_F32_16X16X4_F32
// Phase B: sequential forward (log-sum-exp) recursion over t < lengths[b]
// Phase C: terminal logsumexp + gold path score -> out[b] = logZ - gold
__launch_bounds__(256, 1)
__global__ void crf_nll_fused(const float* __restrict__ feat,   // [B,S,H]
                              const float* __restrict__ Wemb,   // [H,T]
                              const float* __restrict__ bias,   // [T]
                              const float* __restrict__ trans,  // [T,T]
                              const int*   __restrict__ lengths,// [B]
                              const int*   __restrict__ tags,   // [B,S]
                              float*       __restrict__ out)    // [B]
{
  extern __shared__ float smem[];
  float* em    = smem;                 // [S_*T_]  = 24576 floats (96 KB)
  float* tr    = em + S_ * T_;         // [T_*T_]  = 2304 floats
  float* part  = tr + T_ * T_;         // [5*T_]   = 240 floats
  float* red   = part + 5 * T_;        // [256]
  float* alpha = red + 256;            // [T_]
  float* msh   = alpha + T_;           // [4] scratch scalars

  const int b   = blockIdx.x;
  const int tid = threadIdx.x;
  const int lid = tid & 31;            // lane in wave32
  const int wv  = tid >> 5;            // wave id (8 waves)

  // --- cooperative load of transitions into LDS ---
  for (int i = tid; i < T_ * T_; i += 256) tr[i] = trans[i];

  // ================= Phase A: emission GEMM into LDS =================
  const float* Ab = feat + (size_t)b * S_ * H_;
#if HAVE_WMMA_F32
  {
    const int half = lid >> 4;   // which half-wave (K split)
    const int m    = lid & 15;   // A row within tile / B+D column within tile
    for (int mt = wv; mt < S_ / 16; mt += 8) {  // 32 row-tiles over 8 waves
      v8f acc0 = {}; v8f acc1 = {}; v8f acc2 = {};
      const float* Ar = Ab + (size_t)(mt * 16 + m) * H_;
      for (int k0 = 0; k0 < H_; k0 += 4) {
        // A 16x4 f32 frag: lanes 0-15 hold K=k0,k0+1; lanes 16-31 hold K=k0+2,k0+3
        v2f af;
        af.x = Ar[k0 + 2 * half];
        af.y = Ar[k0 + 2 * half + 1];
        // B 4x16 f32 frags (rows striped across lanes, mirrored layout)
        const float* Wr0 = Wemb + (size_t)(k0 + 2 * half) * T_;
        const float* Wr1 = Wr0 + T_;
        v2f bf0; bf0.x = Wr0[m];      bf0.y = Wr1[m];
        v2f bf1; bf1.x = Wr0[16 + m]; bf1.y = Wr1[16 + m];
        v2f bf2; bf2.x = Wr0[32 + m]; bf2.y = Wr1[32 + m];
        acc0 = __builtin_amdgcn_wmma_f32_16x16x4_f32(false, af, false, bf0,
                                                     (short)0, acc0, false, false);
        acc1 = __builtin_amdgcn_wmma_f32_16x16x4_f32(false, af, false, bf1,
                                                     (short)0, acc1, false, false);
        acc2 = __builtin_amdgcn_wmma_f32_16x16x4_f32(false, af, false, bf2,
                                                     (short)0, acc2, false, false);
      }
      // D 16x16 f32 layout: lane gives N = m, VGPR r gives M = r + 8*half
      const float b0 = bias[m];
      const float b1 = bias[16 + m];
      const float b2 = bias[32 + m];
      #pragma unroll
      for (int r = 0; r < 8; ++r) {
        const int srow = mt * 16 + r + 8 * half;
        em[srow * T_ + m]      = acc0[r] + b0;
        em[srow * T_ + 16 + m] = acc1[r] + b1;
        em[srow * T_ + 32 + m] = acc2[r] + b2;
      }
    }
  }
#else
  // scalar fallback (should not be taken on gfx1250)
  for (int o = tid; o < S_ * T_; o += 256) {
    const int s = o / T_, t = o % T_;
    const float* Ar = Ab + (size_t)s * H_;
    float acc = bias[t];
    for (int k = 0; k < H_; ++k) acc += Ar[k] * Wemb[k * T_ + t];
    em[o] = acc;
  }
#endif

  // init alpha
  if (tid < T_) alpha[tid] = (tid == START_IDX) ? 0.0f : NEGV;
  __syncthreads();  // em, tr, alpha ready

  // ================= Phase B: forward recursion =================
  const int len = lengths[b];
  const int g  = tid / T_;          // 0..4 valid, 5 idle
  const int j  = tid % T_;          // 'to' tag
  const int i0 = g * 10;
  const int i1 = (g == 4) ? T_ : (i0 + 10);

  for (int t = 0; t < len; ++t) {
    // wave 0: global max of alpha for stabilization
    if (wv == 0) {
      float v = alpha[lid];
      if (lid < 16) v = fmaxf(v, alpha[32 + lid]);
      #pragma unroll
      for (int off = 16; off > 0; off >>= 1)
        v = fmaxf(v, __shfl_xor(v, off, 32));
      if (lid == 0) msh[0] = v;
    }
    __syncthreads();
    const float mmax = msh[0];
    if (g < 5) {
      float s = 0.0f;
      for (int i = i0; i < i1; ++i)
        s += __expf(alpha[i] + tr[i * T_ + j] - mmax);
      part[g * T_ + j] = s;
    }
    __syncthreads();
    if (tid < T_) {
      const float s = part[j] + part[T_ + j] + part[2 * T_ + j] +
                      part[3 * T_ + j] + part[4 * T_ + j];
      alpha[j] = em[t * T_ + j] + mmax + __logf(s);
    }
    __syncthreads();
  }

  // ================= Phase C: terminal logsumexp + gold =================
  float allp = 0.0f;
  if (wv == 0) {
    float v  = alpha[lid] + tr[lid * T_ + STOP_IDX];
    float v2 = (lid < 16) ? (alpha[32 + lid] + tr[(32 + lid) * T_ + STOP_IDX])
                          : -__builtin_inff();
    float M = fmaxf(v, v2);
    #pragma unroll
    for (int off = 16; off > 0; off >>= 1)
      M = fmaxf(M, __shfl_xor(M, off, 32));
    float s = __expf(v - M) + ((lid < 16) ? __expf(v2 - M) : 0.0f);
    #pragma unroll
    for (int off = 16; off > 0; off >>= 1)
      s += __shfl_xor(s, off, 32);
    allp = M + __logf(s);           // every lane of wave 0 holds logZ
  }

  // gold path score (all 256 threads)
  const int* tgb = tags + (size_t)b * S_;
  float gacc = 0.0f;
  for (int s = tid; s < len; s += 256) {
    const int to   = tgb[s];
    const int from = (s == 0) ? START_IDX : tgb[s - 1];
    gacc += em[s * T_ + to] + tr[from * T_ + to];
  }
  red[tid] = gacc;
  __syncthreads();
  #pragma unroll
  for (int off = 128; off > 0; off >>= 1) {
    if (tid < off) red[tid] += red[tid + off];
    __syncthreads();
  }
  if (tid == 0) {
    const int last = tgb[len - 1];
    const float gold = red[0] + tr[last * T_ + STOP_IDX];
    out[b] = allp - gold;
  }
}

extern "C" void kernel_launch(void* const* d_in, const int* in_sizes, int n_in,
                              void* d_out, int out_size, void* d_ws, size_t ws_size,
                              hipStream_t stream) {
  (void)in_sizes; (void)n_in; (void)out_size; (void)d_ws; (void)ws_size;
  const float* feat = (const float*)d_in[0];   // [B,S,H] f32
  const float* Wemb = (const float*)d_in[1];   // [H,T]   f32
  const float* bias = (const float*)d_in[2];   // [T]     f32
  const float* tr   = (const float*)d_in[3];   // [T,T]   f32
  const int*   len  = (const int*)d_in[4];     // [B]     i32
  const int*   tags = (const int*)d_in[5];     // [B,S]   i32
  float* out = (float*)d_out;                  // [B]     f32

  const size_t shmem =
      (size_t)(S_ * T_ + T_ * T_ + 5 * T_ + 256 + T_ + 4) * sizeof(float);
  crf_nll_fused<<<B_, 256, shmem, stream>>>(feat, Wemb, bias, tr, len, tags, out);
}